// SinkhornDistance_53944789238319
// MI455X (gfx1250) — compile-verified
//
#include <hip/hip_runtime.h>

// CDNA5 / gfx1250: wave32, WMMA 16x16x4 f32.
typedef __attribute__((ext_vector_type(2))) float v2f;
typedef __attribute__((ext_vector_type(8))) float v8f;

// exp(-cost/EPS) = exp2(cost * (-1/EPS * log2(e))), EPS = 0.1
#define KEXP (-14.426950408889634f)

// One block = one (batch b, 16-row i-tile). 256 threads = 8 wave32s.
// Each wave handles j-tiles jt = t*8 + wave (uniform trip count -> scalar
// loop control, EXEC untouched around WMMA). Cross term xs_i*ys_j + x_i*y_j
// computed with v_wmma_f32_16x16x4_f32 (A = [xs|x|0|0], B = [ys;y;0;0]).
__global__ __launch_bounds__(256) void sinkhorn_tile_kernel(
    const float* __restrict__ x,    // [B,N]
    const float* __restrict__ y,    // [B,M]
    const float* __restrict__ xs,   // [B,N] (x_support)
    const float* __restrict__ ys,   // [B,M] (y_support)
    float* __restrict__ partials,   // [B * (N/16)]
    int N, int M)
{
    const int blk    = blockIdx.x;
    const int tilesI = N >> 4;
    const int b      = blk / tilesI;
    const int it     = blk % tilesI;
    const int i0     = it << 4;

    const int tid  = threadIdx.x;
    const int lane = tid & 31;
    const int wave = tid >> 5;
    const int half = lane >> 4;    // 0: lanes 0-15 (K=0,1), 1: lanes 16-31 (K=2,3)
    const int l15  = lane & 15;

    const float* xb  = x  + (size_t)b * N;
    const float* xsb = xs + (size_t)b * N;
    const float* yb  = y  + (size_t)b * M;
    const float* ysb = ys + (size_t)b * M;

    // A operand (16x4 f32): lanes 0-15 carry K=0 (xs) and K=1 (x);
    // lanes 16-31 carry K=2,3 which we zero (rank-2 contraction).
    v2f aop;
    aop.x = (half == 0) ? xsb[i0 + l15] : 0.0f;
    aop.y = (half == 0) ? xb [i0 + l15] : 0.0f;

    // Per-lane row terms sx[M] = xs_M^2 + x_M^2 for the 8 rows this lane
    // sees in the C/D layout: M = r + 8*half.
    float sxr[8];
#pragma unroll
    for (int r = 0; r < 8; ++r) {
        const int m = i0 + r + (half << 3);
        const float s = xsb[m];
        const float v = xb [m];
        sxr[r] = s * s + v * v;
    }

    // 4 accumulators to break the fmac dependence chain.
    float acc0 = 0.0f, acc1 = 0.0f, acc2 = 0.0f, acc3 = 0.0f;

    const int iters = (M >> 4) >> 3;   // j-tiles per wave (uniform)
    for (int t = 0; t < iters; ++t) {
        const int j0 = ((t << 3) + wave) << 4;
        const float bs = ysb[j0 + l15];
        const float bv = yb [j0 + l15];
        const float sy = bs * bs + bv * bv;   // column term for N = lane&15

        // B operand (4x16 f32): lanes 0-15 carry K=0 (ys), K=1 (y);
        // lanes 16-31 (K=2,3) zeroed.
        v2f bop;
        bop.x = (half == 0) ? bs : 0.0f;
        bop.y = (half == 0) ? bv : 0.0f;

        v8f c = {};
        // D = A x B : cross[i][j] = xs_i*ys_j + x_i*y_j  (exact fp32)
        c = __builtin_amdgcn_wmma_f32_16x16x4_f32(
                false, aop, false, bop, (short)0, c, false, false);

#pragma unroll
        for (int r = 0; r < 8; ++r) {
            const float cost = fmaf(-2.0f, c[r], sxr[r] + sy);
            const float p    = __builtin_amdgcn_exp2f(cost * KEXP);
            switch (r & 3) {
                case 0: acc0 = fmaf(p, cost, acc0); break;
                case 1: acc1 = fmaf(p, cost, acc1); break;
                case 2: acc2 = fmaf(p, cost, acc2); break;
                default: acc3 = fmaf(p, cost, acc3); break;
            }
        }
    }

    float acc = (acc0 + acc1) + (acc2 + acc3);

    // wave32 reduction
#pragma unroll
    for (int off = 16; off > 0; off >>= 1)
        acc += __shfl_xor(acc, off, 32);

    __shared__ float wsum[8];
    if (lane == 0) wsum[wave] = acc;
    __syncthreads();
    if (tid == 0) {
        float tsum = 0.0f;
#pragma unroll
        for (int w = 0; w < 8; ++w) tsum += wsum[w];
        partials[blk] = tsum;
    }
}

// Deterministic final reduction: one block per batch sums its tile partials.
__global__ __launch_bounds__(256) void sinkhorn_reduce_kernel(
    const float* __restrict__ partials, float* __restrict__ out, int per_b)
{
    const int b   = blockIdx.x;
    const int tid = threadIdx.x;
    float acc = 0.0f;
    for (int i = tid; i < per_b; i += 256)
        acc += partials[(size_t)b * per_b + i];
#pragma unroll
    for (int off = 16; off > 0; off >>= 1)
        acc += __shfl_xor(acc, off, 32);
    __shared__ float wsum[8];
    const int lane = tid & 31, wave = tid >> 5;
    if (lane == 0) wsum[wave] = acc;
    __syncthreads();
    if (tid == 0) {
        float t = 0.0f;
#pragma unroll
        for (int w = 0; w < 8; ++w) t += wsum[w];
        out[b] = t;
    }
}

extern "C" void kernel_launch(void* const* d_in, const int* in_sizes, int n_in,
                              void* d_out, int out_size, void* d_ws, size_t ws_size,
                              hipStream_t stream) {
    // setup_inputs order: x[B,N], y[B,M], x_support[B,N], y_support[B,M]; out[B].
    const float* x  = (const float*)d_in[0];
    const float* y  = (const float*)d_in[1];
    const float* xs = (const float*)d_in[2];
    const float* ys = (const float*)d_in[3];
    float* out = (float*)d_out;

    const int B = out_size;                 // 8
    const int N = in_sizes[0] / B;          // 4096
    const int M = in_sizes[1] / B;          // 4096
    const int tilesI = N / 16;              // 256
    float* partials = (float*)d_ws;         // B * tilesI floats (8 KB)

    sinkhorn_tile_kernel<<<B * tilesI, 256, 0, stream>>>(x, y, xs, ys, partials, N, M);
    sinkhorn_reduce_kernel<<<B, 256, 0, stream>>>(partials, out, tilesI);
}